// SAGE_11587821765290
// MI455X (gfx1250) — compile-verified
//
#include <hip/hip_runtime.h>
#include <hip/hip_bf16.h>

#define NN 50000
#define EE 800000
#define FIN 128
#define HH 64
#define PP 16
#define CC 40

typedef __attribute__((ext_vector_type(2))) float v2f;
typedef __attribute__((ext_vector_type(8))) float v8f;

// -------- pos embedding: hyp = tanh(pos @ W_pos + b_pos), [N,16]x[16,64] --------
__global__ void pos_embed_kernel(const float* __restrict__ pos,
                                 const float* __restrict__ Wp,
                                 const float* __restrict__ bp,
                                 float* __restrict__ out) {
  int i = blockIdx.x * blockDim.x + threadIdx.x;
  if (i >= NN * HH) return;
  int n = i >> 6, h = i & 63;
  float acc = bp[h];
#pragma unroll
  for (int p = 0; p < PP; ++p) acc = fmaf(pos[n * PP + p], Wp[p * HH + h], acc);
  out[i] = tanhf(acc);
}

// -------- weight transpose: WT[c][k] = W[k][c]  (once per launch, tiny) --------
__global__ void transpose_kernel(const float* __restrict__ W, float* __restrict__ WT,
                                 int K, int C) {
  int i = blockIdx.x * blockDim.x + threadIdx.x;
  if (i >= K * C) return;
  int k = i / C, c = i - k * C;
  WT[(size_t)c * K + k] = W[i];
}

// -------- generic WMMA f32 GEMM (fully unrolled): --------
//   out = [rowinv ⊙] A1 @ W1 [+ A2 @ W2] + bias [+ addv] [relu]
// One wave computes a 16x16 output tile with V_WMMA_F32_16X16X4_F32.
// Lane mapping (wave32): l = lane&15, half = lane>>4.
//   A frag: a[r] = A[row0+l][k0 + 2*half + r]          (contiguous b64 load)
//   B frag: b[r] = WT[col][k0 + 2*half + r]            (contiguous b64 load, W pre-transposed)
//   C/D:    c[r] -> row (row0 + 8*half + r), col (col0 + l)
// Column guard for CCOLS=40 is an address clamp (v_cndmask), no EXEC divergence.
template <int K1, int K2, int CCOLS>
__global__ void gemm_wmma_kernel(const float* __restrict__ A1,
                                 const float* __restrict__ W1T,
                                 const float* __restrict__ A2,
                                 const float* __restrict__ W2T,
                                 const float* __restrict__ bias,
                                 const float* __restrict__ rowinv,  // per-row scale on A1 (mean fuse)
                                 const float* __restrict__ addv,    // optional [N,CCOLS] add
                                 int relu,
                                 float* __restrict__ out) {
  const int lane = threadIdx.x & 31;
  const int wave = threadIdx.x >> 5;
  const int half = lane >> 4;
  const int l    = lane & 15;
  const int row0 = blockIdx.x * 16;
  const int col0 = wave * 16;
  const int col  = col0 + l;
  const int col_ld = (col < CCOLS) ? col : (CCOLS - 1);  // clamp, not branch

  const float s1 = rowinv ? rowinv[row0 + l] : 1.0f;

  v8f c = {};

  const float* a1p = A1  + (size_t)(row0 + l) * K1 + half * 2;
  const float* b1p = W1T + (size_t)col_ld    * K1 + half * 2;
#pragma unroll
  for (int k0 = 0; k0 < K1; k0 += 4) {
    v2f a = *reinterpret_cast<const v2f*>(a1p + k0);
    a.x *= s1;
    a.y *= s1;
    v2f b = *reinterpret_cast<const v2f*>(b1p + k0);
    c = __builtin_amdgcn_wmma_f32_16x16x4_f32(false, a, false, b,
                                              (short)0, c, false, false);
  }

  if (K2 > 0) {
    const float* a2p = A2  + (size_t)(row0 + l) * K2 + half * 2;
    const float* b2p = W2T + (size_t)col_ld    * K2 + half * 2;
#pragma unroll
    for (int k0 = 0; k0 < K2; k0 += 4) {
      v2f a = *reinterpret_cast<const v2f*>(a2p + k0);
      v2f b = *reinterpret_cast<const v2f*>(b2p + k0);
      c = __builtin_amdgcn_wmma_f32_16x16x4_f32(false, a, false, b,
                                                (short)0, c, false, false);
    }
  }

  if (col < CCOLS) {
    const float bv = bias[col];
#pragma unroll
    for (int r = 0; r < 8; ++r) {
      const int row = row0 + half * 8 + r;
      float v = c[r] + bv;
      if (addv) v += addv[(size_t)row * CCOLS + col];
      if (relu) v = fmaxf(v, 0.0f);
      out[(size_t)row * CCOLS + col] = v;
    }
  }
}

// -------- degree counts (static graph: once per launch) --------
__global__ void count_kernel(const int* __restrict__ dst, float* __restrict__ cnt) {
  int e = blockIdx.x * blockDim.x + threadIdx.x;
  if (e >= EE) return;
  atomicAdd(&cnt[dst[e]], 1.0f);
}

// -------- cnt -> 1/clip(cnt,1)  (in place) --------
__global__ void invcnt_kernel(float* __restrict__ cnt) {
  int n = blockIdx.x * blockDim.x + threadIdx.x;
  if (n >= NN) return;
  cnt[n] = 1.0f / fmaxf(cnt[n], 1.0f);
}

// -------- scatter-add of messages: agg[dst] += h[src], float4 per thread --------
__global__ void scatter_kernel(const int* __restrict__ src, const int* __restrict__ dst,
                               const float* __restrict__ h, float* __restrict__ agg) {
  int i = blockIdx.x * blockDim.x + threadIdx.x;
  if (i >= EE * 16) return;
  int e = i >> 4;
  int f = (i & 15) << 2;
  int s = src[e], d = dst[e];
  const float4 v = *reinterpret_cast<const float4*>(h + (size_t)s * HH + f);
  float* o = agg + (size_t)d * HH + f;
  atomicAdd(o + 0, v.x);
  atomicAdd(o + 1, v.y);
  atomicAdd(o + 2, v.z);
  atomicAdd(o + 3, v.w);
}

// -------- row-wise log_softmax over C=40 --------
__global__ void logsoftmax_kernel(const float* __restrict__ emb, float* __restrict__ out) {
  int n = blockIdx.x * blockDim.x + threadIdx.x;
  if (n >= NN) return;
  float v[CC];
  float m = -3.0e38f;
#pragma unroll
  for (int c = 0; c < CC; ++c) {
    v[c] = emb[(size_t)n * CC + c];
    m = fmaxf(m, v[c]);
  }
  float s = 0.0f;
#pragma unroll
  for (int c = 0; c < CC; ++c) s += expf(v[c] - m);
  const float lse = m + logf(s);
#pragma unroll
  for (int c = 0; c < CC; ++c) out[(size_t)n * CC + c] = v[c] - lse;
}

extern "C" void kernel_launch(void* const* d_in, const int* in_sizes, int n_in,
                              void* d_out, int out_size, void* d_ws, size_t ws_size,
                              hipStream_t stream) {
  const float* x_h    = (const float*)d_in[0];
  const float* pos    = (const float*)d_in[1];
  const int*   ei     = (const int*)d_in[2];
  const float* W_pos  = (const float*)d_in[3];
  const float* b_pos  = (const float*)d_in[4];
  const float* W_init = (const float*)d_in[5];
  const float* b_init = (const float*)d_in[6];
  const float* W_l    = (const float*)d_in[7];
  const float* W_r    = (const float*)d_in[8];
  const float* b_conv = (const float*)d_in[9];
  const float* W_last = (const float*)d_in[10];
  const float* b_last = (const float*)d_in[11];
  const int* src = ei;        // edge_index[0]
  const int* dst = ei + EE;   // edge_index[1]

  float* wsf = (float*)d_ws;
  float* hyp = wsf;                      // N*H
  float* h   = hyp + (size_t)NN * HH;    // N*H
  float* agg = h   + (size_t)NN * HH;    // N*H
  float* cnt = agg + (size_t)NN * HH;    // N   (becomes 1/deg after invcnt)
  float* wT  = cnt + NN;                 // transposed weights below
  float* W_initT = wT;                        // 64*128
  float* WlT     = W_initT + 64 * 128;        // 3 * 64*64
  float* WrT     = WlT + 3 * 64 * 64;         // 3 * 64*64
  float* W_lastT = WrT + 3 * 64 * 64;         // 40*64

  float* emb = (float*)d_out;            // N*C (output 0)
  float* lsm = emb + (size_t)NN * CC;    // N*C (output 1)

  // 0) one-time weight transposes (tiny)
  transpose_kernel<<<(FIN * HH + 255) / 256, 256, 0, stream>>>(W_init, W_initT, FIN, HH);
  for (int i = 0; i < 3; ++i) {
    transpose_kernel<<<(HH * HH + 255) / 256, 256, 0, stream>>>(
        W_l + (size_t)i * HH * HH, WlT + (size_t)i * HH * HH, HH, HH);
    transpose_kernel<<<(HH * HH + 255) / 256, 256, 0, stream>>>(
        W_r + (size_t)i * HH * HH, WrT + (size_t)i * HH * HH, HH, HH);
  }
  transpose_kernel<<<(HH * CC + 255) / 256, 256, 0, stream>>>(W_last, W_lastT, HH, CC);

  // 1) positional embedding
  pos_embed_kernel<<<(NN * HH + 255) / 256, 256, 0, stream>>>(pos, W_pos, b_pos, hyp);

  // 2) h = x_h @ W_init + b_init   (K=128, 4 col-tiles)
  gemm_wmma_kernel<FIN, 0, HH><<<NN / 16, 32 * (HH / 16), 0, stream>>>(
      x_h, W_initT, nullptr, nullptr, b_init, nullptr, nullptr, 0, h);

  // 3) degree counts -> reciprocal (static graph: once)
  hipMemsetAsync(cnt, 0, NN * sizeof(float), stream);
  count_kernel<<<(EE + 255) / 256, 256, 0, stream>>>(dst, cnt);
  invcnt_kernel<<<(NN + 255) / 256, 256, 0, stream>>>(cnt);

  // 4) three SAGE layers: h = (agg/deg)@Wl + h@Wr + b [+ hyp, relu]
  for (int i = 0; i < 3; ++i) {
    hipMemsetAsync(agg, 0, (size_t)NN * HH * sizeof(float), stream);
    scatter_kernel<<<(EE * 16 + 255) / 256, 256, 0, stream>>>(src, dst, h, agg);
    const float* addv = (i != 2) ? hyp : nullptr;
    gemm_wmma_kernel<HH, HH, HH><<<NN / 16, 32 * (HH / 16), 0, stream>>>(
        agg, WlT + (size_t)i * HH * HH,
        h,   WrT + (size_t)i * HH * HH,
        b_conv + (size_t)i * HH, cnt, addv, (i != 2) ? 1 : 0, h);
  }

  // 5) emb = h @ W_last + b_last  (C=40 -> 3 col-tiles, clamped)
  gemm_wmma_kernel<HH, 0, CC><<<NN / 16, 32 * ((CC + 15) / 16), 0, stream>>>(
      h, W_lastT, nullptr, nullptr, b_last, nullptr, nullptr, 0, emb);

  // 6) log_softmax
  logsoftmax_kernel<<<(NN + 255) / 256, 256, 0, stream>>>(emb, lsm);
}